// Model_54992761258561
// MI455X (gfx1250) — compile-verified
//
#include <hip/hip_runtime.h>
#include <hip/hip_bf16.h>
#include <math.h>

// ---------------------------------------------------------------------------
// Model dims (match reference)
// ---------------------------------------------------------------------------
#define NNODES 20000
#define NEDGES 320000
#define NHEADS 2
#define CDIM   128      // HID == GOUT == 128
#define NATTR  256
#define NPAIR  16384
#define SLOPE_F 0.2f
#define BN_EPS  1e-5f

typedef float v2f __attribute__((ext_vector_type(2)));
typedef float v8f __attribute__((ext_vector_type(8)));

// ---------------------------------------------------------------------------
// float atomic max via signed/unsigned bit trick
// ---------------------------------------------------------------------------
__device__ __forceinline__ void atomicMaxF(float* addr, float v) {
    if (v >= 0.0f) atomicMax((int*)addr, __float_as_int(v));
    else           atomicMin((unsigned int*)addr, __float_as_uint(v));
}

__device__ __forceinline__ v8f wmma4(v2f a, v2f b, v8f c) {
    return __builtin_amdgcn_wmma_f32_16x16x4_f32(
        false, a, false, b, (short)0, c, false, false);
}

// ---------------------------------------------------------------------------
// fp32 WMMA GEMM:  C[M,Nc] = A[M,K] @ B[K,Nc] (+bias)(+relu)(+BN epilogue)
// One wave -> 32x64 strip (2 row sub-tiles x 4 col tiles), K stepped by 4.
// Requires M % 32 == 0 and K % 4 == 0 (true for all call sites).
// A-frag (ISA 7.12.2): lanes 0-15 hold {K=k0,k0+1} of row M=lane,
//                      lanes 16-31 hold {K=k0+2,k0+3} of row M=lane-16.
// B-frag: lanes 0-15 rows k0,k0+1 at col=lane; lanes 16-31 rows k0+2,k0+3.
// C/D: VGPR v, lanes 0-15 -> row v, lanes 16-31 -> row v+8; col = lane & 15.
// B fragments are shared by both row sub-tiles (2 WMMAs per B load pair).
// ---------------------------------------------------------------------------
__global__ __launch_bounds__(128) void wmma_gemm_f32(
    const float* __restrict__ A, const float* __restrict__ B,
    const float* __restrict__ bias, float* __restrict__ C,
    int M, int K, int Nc, int relu,
    const float* __restrict__ bng, const float* __restrict__ bnb,
    const float* __restrict__ bnrm, const float* __restrict__ bnrv)
{
    const int lane  = threadIdx.x & 31;
    const int wave  = threadIdx.x >> 5;
    const int row0  = blockIdx.x * 128 + wave * 32;
    const int col0  = blockIdx.y * 64;
    if (row0 >= M) return;                 // wave-uniform: EXEC stays all-1 for WMMA

    const int mlane = lane & 15;
    const int khalf = (lane >> 4) << 1;    // 0 or 2

    v8f acc[2][4] = {};
    const float* arow0 = A + (size_t)(row0 + mlane) * K + khalf;
    const float* arow1 = arow0 + (size_t)16 * K;

    if (col0 + 64 <= Nc) {
        // ---- fast path: all 4 col tiles fully in bounds, no masking ----
        const float* bp = B + col0 + mlane;
        for (int k0 = 0; k0 < K; k0 += 4) {
            __builtin_prefetch(arow0 + k0 + 64, 0, 1);   // global_prefetch_b8
            v2f a0, a1;
            a0.x = arow0[k0]; a0.y = arow0[k0 + 1];
            a1.x = arow1[k0]; a1.y = arow1[k0 + 1];
            const float* b0 = bp + (size_t)(k0 + khalf) * Nc;
            const float* b1 = b0 + Nc;
            v2f bf[4];
#pragma unroll
            for (int t = 0; t < 4; ++t) {
                bf[t].x = b0[t * 16];
                bf[t].y = b1[t * 16];
            }
#pragma unroll
            for (int t = 0; t < 4; ++t) {
                acc[0][t] = wmma4(a0, bf[t], acc[0][t]);
                acc[1][t] = wmma4(a1, bf[t], acc[1][t]);
            }
        }
    } else {
        // ---- partial-column path: clamped-address load + value select ----
        for (int k0 = 0; k0 < K; k0 += 4) {
            v2f a0, a1;
            a0.x = arow0[k0]; a0.y = arow0[k0 + 1];
            a1.x = arow1[k0]; a1.y = arow1[k0 + 1];
            const float* b0 = B + (size_t)(k0 + khalf) * Nc;
            const float* b1 = b0 + Nc;
            v2f bf[4];
#pragma unroll
            for (int t = 0; t < 4; ++t) {
                const int col  = col0 + t * 16 + mlane;
                const int colc = (col < Nc) ? col : (Nc - 1);
                const float vx = b0[colc];
                const float vy = b1[colc];
                bf[t].x = (col < Nc) ? vx : 0.0f;
                bf[t].y = (col < Nc) ? vy : 0.0f;
            }
#pragma unroll
            for (int t = 0; t < 4; ++t) {
                acc[0][t] = wmma4(a0, bf[t], acc[0][t]);
                acc[1][t] = wmma4(a1, bf[t], acc[1][t]);
            }
        }
    }

#pragma unroll
    for (int t = 0; t < 4; ++t) {
        const int col = col0 + t * 16 + mlane;
        if (col >= Nc) continue;
        const float bsv = bias ? bias[col] : 0.0f;
        float g = 1.0f, bb = 0.0f, rm = 0.0f, rinv = 1.0f;
        if (bng) {
            g    = bng[col];
            bb   = bnb[col];
            rm   = bnrm[col];
            rinv = rsqrtf(bnrv[col] + BN_EPS);
        }
#pragma unroll
        for (int r = 0; r < 2; ++r) {
            const int rbase = row0 + r * 16 + ((lane >> 4) << 3);
#pragma unroll
            for (int v = 0; v < 8; ++v) {
                float val = acc[r][t][v] + bsv;
                if (relu) val = fmaxf(val, 0.0f);
                if (bng)  val = (val - rm) * rinv * g + bb;
                C[(size_t)(rbase + v) * Nc + col] = val;
            }
        }
    }
}

// ---------------------------------------------------------------------------
// es[n,h] = <H[n,h,:], a_src[h,:]> ; ed likewise. One wave per (n,h).
// ---------------------------------------------------------------------------
__global__ __launch_bounds__(256) void gat_scores(
    const float* __restrict__ H, const float* __restrict__ asrc,
    const float* __restrict__ adst, float* __restrict__ es,
    float* __restrict__ ed, int Nn)
{
    const int gw   = (int)((blockIdx.x * blockDim.x + threadIdx.x) >> 5);
    const int lane = threadIdx.x & 31;
    if (gw >= Nn * NHEADS) return;
    const int n = gw >> 1, h = gw & 1;
    const float* hp = H + (size_t)n * NHEADS * CDIM + h * CDIM;
    const float* as = asrc + h * CDIM;
    const float* ad = adst + h * CDIM;
    float s = 0.0f, d = 0.0f;
    for (int c = lane; c < CDIM; c += 32) {
        const float hv = hp[c];
        s += hv * as[c];
        d += hv * ad[c];
    }
    for (int off = 16; off; off >>= 1) {
        s += __shfl_down(s, off);
        d += __shfl_down(d, off);
    }
    if (lane == 0) { es[gw] = s; ed[gw] = d; }
}

__global__ void fill_f32(float* __restrict__ p, float v, long n)
{
    const long i = (long)blockIdx.x * blockDim.x + threadIdx.x;
    if (i < n) p[i] = v;
}

// edge pass 1: e = leaky_relu(es[src]+ed[dst]); segment max into m[dst]
__global__ void edge_logits_max(
    const int* __restrict__ ei, const float* __restrict__ es,
    const float* __restrict__ ed, float* __restrict__ ebuf,
    float* __restrict__ mx, int E, int Nn)
{
    const int idx = blockIdx.x * blockDim.x + threadIdx.x;
    const int total = (E + Nn) * NHEADS;
    if (idx >= total) return;
    const int e = idx >> 1, h = idx & 1;
    const int s = (e < E) ? ei[e]     : (e - E);   // row 0 of edge_index = src
    const int d = (e < E) ? ei[E + e] : (e - E);   // row 1 = dst
    float v = es[s * 2 + h] + ed[d * 2 + h];
    v = (v > 0.0f) ? v : SLOPE_F * v;
    ebuf[idx] = v;
    atomicMaxF(&mx[d * 2 + h], v);
}

// edge pass 2: ex = exp(e - m[dst]); segment sum into s[dst]
__global__ void edge_expsum(
    const int* __restrict__ ei, float* __restrict__ ebuf,
    const float* __restrict__ mx, float* __restrict__ ssum, int E, int Nn)
{
    const int idx = blockIdx.x * blockDim.x + threadIdx.x;
    const int total = (E + Nn) * NHEADS;
    if (idx >= total) return;
    const int e = idx >> 1, h = idx & 1;
    const int d = (e < E) ? ei[E + e] : (e - E);
    const float v = expf(ebuf[idx] - mx[d * 2 + h]);
    ebuf[idx] = v;
    atomicAdd(&ssum[d * 2 + h], v);
}

// edge pass 3: agg[dst,h,:] += H[src,h,:] * (ex / s[dst,h]). One wave per (edge,h).
__global__ __launch_bounds__(256) void edge_aggregate(
    const int* __restrict__ ei, const float* __restrict__ ebuf,
    const float* __restrict__ ssum, const float* __restrict__ H,
    float* __restrict__ agg, int E, int Nn)
{
    const int gw   = (int)((blockIdx.x * blockDim.x + threadIdx.x) >> 5);
    const int lane = threadIdx.x & 31;
    const int total = (E + Nn) * NHEADS;
    if (gw >= total) return;
    const int e = gw >> 1, h = gw & 1;
    const int s = (e < E) ? ei[e]     : (e - E);
    const int d = (e < E) ? ei[E + e] : (e - E);
    const float alpha = ebuf[gw] / ssum[d * 2 + h];
    const float* hp = H + (size_t)s * NHEADS * CDIM + h * CDIM;
    float* op = agg + (size_t)d * NHEADS * CDIM + h * CDIM;
    for (int c = lane; c < CDIM; c += 32)
        atomicAdd(&op[c], hp[c] * alpha);
}

// out[n,c] = mean over heads + bias
__global__ void head_mean_bias(
    const float* __restrict__ agg, const float* __restrict__ b,
    float* __restrict__ outp, int Nn)
{
    const int idx = blockIdx.x * blockDim.x + threadIdx.x;
    if (idx >= Nn * CDIM) return;
    const int n = idx >> 7, c = idx & 127;
    outp[idx] = 0.5f * (agg[(size_t)n * 2 * CDIM + c] +
                        agg[(size_t)n * 2 * CDIM + CDIM + c]) + b[c];
}

// Xp[r, 0:512] = [topo[i0], attr[i0], topo[i1], attr[i1]]
__global__ void gather_pairs(
    const float* __restrict__ topo, const float* __restrict__ attr,
    const int* __restrict__ xp, float* __restrict__ Xp, int rows, int p0)
{
    const int idx = blockIdx.x * blockDim.x + threadIdx.x;
    if (idx >= rows * 512) return;
    const int r = idx >> 9, c = idx & 511;
    const int p = p0 + r;
    const int node = (c < 256) ? xp[p * 2] : xp[p * 2 + 1];
    const int cc = c & 255;
    const float v = (cc < 128) ? topo[(size_t)node * 128 + cc]
                               : attr[(size_t)node * 128 + (cc - 128)];
    Xp[(size_t)r * 512 + c] = v;
}

// out[p] = sigmoid(<h2[r,:256], w3> + b3). One wave per row.
__global__ __launch_bounds__(256) void logits_kernel(
    const float* __restrict__ h2, const float* __restrict__ w3,
    const float* __restrict__ b3, float* __restrict__ out, int rows, int p0)
{
    const int gw   = (int)((blockIdx.x * blockDim.x + threadIdx.x) >> 5);
    const int lane = threadIdx.x & 31;
    if (gw >= rows) return;
    const float* hr = h2 + (size_t)gw * 256;
    float s = 0.0f;
    for (int c = lane; c < 256; c += 32) s += hr[c] * w3[c];
    for (int off = 16; off; off >>= 1) s += __shfl_down(s, off);
    if (lane == 0) out[p0 + gw] = 1.0f / (1.0f + expf(-(s + b3[0])));
}

// ---------------------------------------------------------------------------
// Host launcher
// ---------------------------------------------------------------------------
extern "C" void kernel_launch(void* const* d_in, const int* in_sizes, int n_in,
                              void* d_out, int out_size, void* d_ws, size_t ws_size,
                              hipStream_t stream)
{
    const int*   edge_index = (const int*)  d_in[0];
    const float* attr_mtx   = (const float*)d_in[1];
    const int*   xpairs     = (const int*)  d_in[2];
    const float* X_feat     = (const float*)d_in[3];
    const float* W1 = (const float*)d_in[4],  *att_src1 = (const float*)d_in[5],
               *att_dst1 = (const float*)d_in[6],  *b1 = (const float*)d_in[7];
    const float* W2 = (const float*)d_in[8],  *att_src2 = (const float*)d_in[9],
               *att_dst2 = (const float*)d_in[10], *b2 = (const float*)d_in[11];
    const float* gt_w1  = (const float*)d_in[12], *gt_b1  = (const float*)d_in[13];
    const float* gt_w2  = (const float*)d_in[14], *gt_b2  = (const float*)d_in[15];
    const float* t2a_w1 = (const float*)d_in[16], *t2a_b1 = (const float*)d_in[17];
    const float* t2a_w2 = (const float*)d_in[18], *t2a_b2 = (const float*)d_in[19];
    const float* ga_w1  = (const float*)d_in[20], *ga_b1  = (const float*)d_in[21];
    const float* ga_w2  = (const float*)d_in[22], *ga_b2  = (const float*)d_in[23];
    const float* a2t_w1 = (const float*)d_in[24], *a2t_b1 = (const float*)d_in[25];
    const float* a2t_w2 = (const float*)d_in[26], *a2t_b2 = (const float*)d_in[27];
    const float* c_w1 = (const float*)d_in[28], *c_b1 = (const float*)d_in[29];
    const float* bn1_g = (const float*)d_in[30], *bn1_b = (const float*)d_in[31];
    const float* bn1_rm = (const float*)d_in[32], *bn1_rv = (const float*)d_in[33];
    const float* c_w2 = (const float*)d_in[34], *c_b2 = (const float*)d_in[35];
    const float* bn2_g = (const float*)d_in[36], *bn2_b = (const float*)d_in[37];
    const float* bn2_rm = (const float*)d_in[38], *bn2_rv = (const float*)d_in[39];
    const float* c_w3 = (const float*)d_in[40], *c_b3 = (const float*)d_in[41];

    float* out = (float*)d_out;
    float* ws  = (float*)d_ws;

    // ---- workspace layout (floats); total 20,760,000 floats = 83.0 MB ----
    const size_t oH    = 0;                                    // [N, 2*C] GAT hidden
    const size_t oAgg  = oH    + (size_t)NNODES * 2 * CDIM;    // [N, 2, C]
    const size_t oHn   = oAgg  + (size_t)NNODES * 2 * CDIM;    // [N, C] layer1 out
    const size_t oEs   = oHn   + (size_t)NNODES * CDIM;
    const size_t oEd   = oEs   + (size_t)NNODES * 2;
    const size_t oM    = oEd   + (size_t)NNODES * 2;
    const size_t oS    = oM    + (size_t)NNODES * 2;
    const size_t oEbuf = oS    + (size_t)NNODES * 2;           // [(E+N)*2]
    const size_t oT100 = oEbuf + (size_t)(NEDGES + NNODES) * 2;// [N, 100]
    const size_t oTopo = oT100 + (size_t)NNODES * 100;         // [N, 128]
    const size_t oAttr = oTopo + (size_t)NNODES * CDIM;        // [N, 128]
    // classifier chunk buffers reuse the (dead) H/agg region:
    const size_t oXp   = 0;                                    // [4096, 512]
    const size_t oH1c  = oXp   + (size_t)4096 * 512;           // [4096, 512]
    const size_t oH2c  = oH1c  + (size_t)4096 * 512;           // [4096, 256]

    auto gemm = [&](const float* A, const float* B, const float* bias, float* Cp,
                    int M, int K, int Ncol, int relu,
                    const float* g = nullptr, const float* bb = nullptr,
                    const float* rm = nullptr, const float* rv = nullptr) {
        dim3 grid((M + 127) / 128, (Ncol + 63) / 64);
        wmma_gemm_f32<<<grid, 128, 0, stream>>>(A, B, bias, Cp, M, K, Ncol,
                                                relu, g, bb, rm, rv);
    };

    const int etot = (NEDGES + NNODES) * 2;  // 680,000 (edge,head) items

    auto run_gat = [&](const float* X, int K, const float* W, const float* asrc,
                       const float* adst, const float* bias, float* outp) {
        gemm(X, W, nullptr, ws + oH, NNODES, K, 2 * CDIM, 0);      // H = X @ W
        gat_scores<<<(NNODES * 2 * 32 + 255) / 256, 256, 0, stream>>>(
            ws + oH, asrc, adst, ws + oEs, ws + oEd, NNODES);
        fill_f32<<<(NNODES * 2 + 255) / 256, 256, 0, stream>>>(
            ws + oM, -__builtin_inff(), NNODES * 2);
        fill_f32<<<(NNODES * 2 + 255) / 256, 256, 0, stream>>>(
            ws + oS, 0.0f, NNODES * 2);
        fill_f32<<<((long)NNODES * 2 * CDIM + 255) / 256, 256, 0, stream>>>(
            ws + oAgg, 0.0f, (long)NNODES * 2 * CDIM);
        edge_logits_max<<<(etot + 255) / 256, 256, 0, stream>>>(
            edge_index, ws + oEs, ws + oEd, ws + oEbuf, ws + oM, NEDGES, NNODES);
        edge_expsum<<<(etot + 255) / 256, 256, 0, stream>>>(
            edge_index, ws + oEbuf, ws + oM, ws + oS, NEDGES, NNODES);
        edge_aggregate<<<((long)etot * 32 + 255) / 256, 256, 0, stream>>>(
            edge_index, ws + oEbuf, ws + oS, ws + oH, ws + oAgg, NEDGES, NNODES);
        head_mean_bias<<<(NNODES * CDIM + 255) / 256, 256, 0, stream>>>(
            ws + oAgg, bias, outp, NNODES);
    };

    // ---- output layout: [logits 16384 | gcn_out N*128 | t2a N*256 | a2t N*128] ----
    float* gcn_out = out + NPAIR;
    float* t2a_out = gcn_out + (size_t)NNODES * CDIM;
    float* a2t_out = t2a_out + (size_t)NNODES * NATTR;

    // ---- GAT encoder ----
    run_gat(X_feat, 64, W1, att_src1, att_dst1, b1, ws + oHn);
    run_gat(ws + oHn, CDIM, W2, att_src2, att_dst2, b2, gcn_out);

    // ---- node MLPs / generators (no activations, per reference) ----
    gemm(gcn_out,    gt_w1,  gt_b1,  ws + oT100, NNODES, 128, 100, 0);
    gemm(ws + oT100, gt_w2,  gt_b2,  ws + oTopo, NNODES, 100, 128, 0);
    gemm(ws + oTopo, t2a_w1, t2a_b1, ws + oT100, NNODES, 128, 100, 0);
    gemm(ws + oT100, t2a_w2, t2a_b2, t2a_out,    NNODES, 100, 256, 0);
    gemm(attr_mtx,   ga_w1,  ga_b1,  ws + oT100, NNODES, 256, 100, 0);
    gemm(ws + oT100, ga_w2,  ga_b2,  ws + oAttr, NNODES, 100, 128, 0);
    gemm(ws + oAttr, a2t_w1, a2t_b1, ws + oT100, NNODES, 128, 100, 0);
    gemm(ws + oT100, a2t_w2, a2t_b2, a2t_out,    NNODES, 100, 128, 0);

    // ---- classifier over pair chunks (relu + eval-mode BN folded into GEMM) ----
    const int CH = 4096;
    for (int p0 = 0; p0 < NPAIR; p0 += CH) {
        gather_pairs<<<(CH * 512 + 255) / 256, 256, 0, stream>>>(
            ws + oTopo, ws + oAttr, xpairs, ws + oXp, CH, p0);
        gemm(ws + oXp,  c_w1, c_b1, ws + oH1c, CH, 512, 512, 1,
             bn1_g, bn1_b, bn1_rm, bn1_rv);
        gemm(ws + oH1c, c_w2, c_b2, ws + oH2c, CH, 512, 256, 1,
             bn2_g, bn2_b, bn2_rm, bn2_rv);
        logits_kernel<<<(CH * 32 + 255) / 256, 256, 0, stream>>>(
            ws + oH2c, c_w3, c_b3, out, CH, p0);
    }
}